// RNNScratch_6459630813260
// MI455X (gfx1250) — compile-verified
//
#include <hip/hip_runtime.h>
#include <hip/hip_bf16.h>
#include <math.h>

// ---------------------------------------------------------------------------
// RNN scratch for MI455X (gfx1250, wave32, WMMA).
//   xw      = X @ W_xh + b            (big GEMM, register-blocked 32x64/wave)
//   state_t = tanh(xw_t + state_{t-1} @ W_hh)  (persistent kernel, W_hh slice
//             in LDS via async-to-LDS, device barrier per step)
// All matrix math uses V_WMMA_F32_16X16X4_F32 (full f32 precision).
// ---------------------------------------------------------------------------

typedef float v2f __attribute__((ext_vector_type(2)));
typedef float v8f __attribute__((ext_vector_type(8)));

#define WAVES_PER_BLOCK 8
#define THREADS_PER_BLOCK (WAVES_PER_BLOCK * 32)

#define REC_BLOCKS 32          // persistent recurrence workgroups
#define SB_DWORDS (512 * 64)   // 1024x32 f32 W_hh slice, pair-interleaved: 128KB

#if __has_builtin(__builtin_amdgcn_global_load_async_to_lds_b32) && \
    __has_builtin(__builtin_amdgcn_s_wait_asynccnt)
#define USE_ASYNC_LDS 1
typedef __attribute__((address_space(1))) int* gptr_i32;
typedef __attribute__((address_space(3))) int* lptr_i32;
#endif

#if __has_builtin(__builtin_amdgcn_tanhf)
#define TANHF(x) __builtin_amdgcn_tanhf(x)   // V_TANH_F32 trans op
#else
#define TANHF(x) tanhf(x)
#endif

__device__ __forceinline__ v8f wmma4(v2f a, v2f b, v8f c) {
    return __builtin_amdgcn_wmma_f32_16x16x4_f32(
        /*neg_a=*/false, a, /*neg_b=*/false, b,
        /*c_mod=*/(short)0, c, /*reuse_a=*/false, /*reuse_b=*/false);
}

// ---------------------------------------------------------------------------
// Fragment layouts (ISA 7.12.2, f32 16x16x4):
//   A (16x4 MxK): lane l holds m=l%16, k=(l/16)*2+{0,1} -> float2
//   B (4x16 KxN): lane l holds n=l%16, k=(l/16)*2+{0,1}
//   D (16x16):    vgpr v, lane l -> m=v+8*(l/16), n=l%16
// ---------------------------------------------------------------------------

// One 16x16 tile per wave, fully streamed (fallback step kernel only).
__device__ __forceinline__ v8f wmma_tile_f32(const float* __restrict__ A,
                                             const float* __restrict__ W,
                                             int row0, int col0,
                                             int K, int N, int lane) {
    const int ln = lane & 15;
    const int hi = lane >> 4;
    v8f acc = {0.f, 0.f, 0.f, 0.f, 0.f, 0.f, 0.f, 0.f};

    const float* __restrict__ arow = A + (size_t)(row0 + ln) * K + 2 * hi;
    const float* __restrict__ bbase = W + (size_t)(2 * hi) * N + col0 + ln;

#pragma unroll 8
    for (int k0 = 0; k0 < K; k0 += 4) {
        v2f a = *(const v2f*)(arow + k0);
        v2f b;
        b.x = bbase[(size_t)k0 * N];
        b.y = bbase[(size_t)(k0 + 1) * N];
        acc = wmma4(a, b, acc);
    }
    return acc;
}

// ---------------------------------------------------------------------------
// Kernel A: Y[M,N] = X[M,K] @ W[K,N] + bias[N]
// Register-blocked: each wave owns a 32x64 macro-tile (2 M-frags x 4 N-frags,
// 8 accumulators). Per K=4 step: 2 A loads + 4 B loads feed 8 WMMAs
// (1.5 dwords/lane/WMMA instead of 4 -> ~2.7x less L0/L2 traffic).
// Waves of a block cover consecutive N macro-tiles of one 32-row A panel.
// ---------------------------------------------------------------------------
__global__ __launch_bounds__(THREADS_PER_BLOCK)
void xproj_kernel(const float* __restrict__ X, const float* __restrict__ W,
                  const float* __restrict__ bias, float* __restrict__ Y,
                  int M, int N, int K) {
    const int lane = threadIdx.x & 31;
    const int wave = threadIdx.x >> 5;
    const int tileId = blockIdx.x * WAVES_PER_BLOCK + wave;
    const int tilesN = N >> 6;          // 64-col macro-tiles
    const int tm = tileId / tilesN;
    const int tn = tileId - tm * tilesN;
    if ((tm << 5) >= M) return;         // wave-uniform

    const int row0 = tm << 5;           // 32 rows
    const int col0 = tn << 6;           // 64 cols
    const int ln = lane & 15;
    const int hi = lane >> 4;

    v8f acc[2][4];
#pragma unroll
    for (int i = 0; i < 2; ++i)
#pragma unroll
        for (int j = 0; j < 4; ++j)
            acc[i][j] = (v8f){0.f, 0.f, 0.f, 0.f, 0.f, 0.f, 0.f, 0.f};

    const float* __restrict__ arow0 = X + (size_t)(row0 + ln) * K + 2 * hi;
    const float* __restrict__ arow1 = arow0 + (size_t)16 * K;
    const float* __restrict__ bbase = W + (size_t)(2 * hi) * N + col0 + ln;

#pragma unroll 2
    for (int k0 = 0; k0 < K; k0 += 4) {
        v2f a0 = *(const v2f*)(arow0 + k0);
        v2f a1 = *(const v2f*)(arow1 + k0);
        v2f b[4];
#pragma unroll
        for (int j = 0; j < 4; ++j) {
            b[j].x = bbase[(size_t)k0 * N + 16 * j];
            b[j].y = bbase[(size_t)(k0 + 1) * N + 16 * j];
        }
#pragma unroll
        for (int j = 0; j < 4; ++j) {
            acc[0][j] = wmma4(a0, b[j], acc[0][j]);
            acc[1][j] = wmma4(a1, b[j], acc[1][j]);
        }
    }

    float bv[4];
#pragma unroll
    for (int j = 0; j < 4; ++j) bv[j] = bias[col0 + 16 * j + ln];

#pragma unroll
    for (int i = 0; i < 2; ++i) {
#pragma unroll
        for (int v = 0; v < 8; ++v) {
            const int m = 16 * i + v + 8 * hi;
            float* __restrict__ yrow = Y + (size_t)(row0 + m) * N + col0 + ln;
#pragma unroll
            for (int j = 0; j < 4; ++j) {
                yrow[16 * j] = acc[i][j][v] + bv[j];
            }
        }
    }
}

// ---------------------------------------------------------------------------
// Persistent recurrence kernel: 32 blocks, each owns a 32-column slice of
// W_hh in LDS (128KB, pair-interleaved) and 8 tiles (4 M-tiles x 2 N-tiles,
// one per wave). Per step: GEMM from LDS, fused tanh epilogue in-place into
// out[t], then a device-wide monotonic-counter barrier.
// ---------------------------------------------------------------------------
__global__ __launch_bounds__(THREADS_PER_BLOCK)
void rnn_recurrence_kernel(const float* __restrict__ W,   // W_hh [H,H]
                           float* __restrict__ out,       // [T,B,H], xw staged
                           unsigned* __restrict__ counter, int T) {
    const int H = 1024, Bz = 64;
    __shared__ float sB[SB_DWORDS];

    const int lane = threadIdx.x & 31;
    const int wave = threadIdx.x >> 5;
    const int ln = lane & 15;
    const int hi = lane >> 4;
    const int col_base = blockIdx.x * 32;
    const int row0 = (wave & 3) << 4;   // 4 M-tiles (B=64)
    const int tn = wave >> 2;           // 2 N-tiles within the slice
    const int col0 = col_base + (tn << 4);

    // ---- One-time preload of W_hh[:, col_base:col_base+32) into LDS.
    // Layout: pair p=k/2, dword offset p*64 + n*2 + (k&1)  -> B fragment is
    // a single ds_load_b64 per lane.
    for (int i = threadIdx.x; i < SB_DWORDS; i += THREADS_PER_BLOCK) {
        const int k = i >> 5;
        const int n = i & 31;
        const int dst = ((k >> 1) << 6) + (n << 1) + (k & 1);
        const float* src = W + (size_t)k * H + col_base + n;
#ifdef USE_ASYNC_LDS
        __builtin_amdgcn_global_load_async_to_lds_b32(
            (gptr_i32)src, (lptr_i32)&sB[dst], 0, 0);
#else
        sB[dst] = *src;
#endif
    }
#ifdef USE_ASYNC_LDS
    __builtin_amdgcn_s_wait_asynccnt(0);
#endif
    __syncthreads();

    const v2f* __restrict__ bp = (const v2f*)sB + (tn << 4) + ln;

    for (int t = 0; t < T; ++t) {
        float* xw = out + (size_t)t * Bz * H;

        v8f acc = {0.f, 0.f, 0.f, 0.f, 0.f, 0.f, 0.f, 0.f};
        if (t > 0) {  // uniform branch; EXEC all-ones around WMMA
            const float* S = xw - (size_t)Bz * H;   // previous state
            const float* __restrict__ arow = S + (size_t)(row0 + ln) * H + 2 * hi;
            // 4 independent accumulator chains (depth 64 instead of 256).
            v8f a0 = acc, a1 = acc, a2 = acc, a3 = acc;
#pragma unroll 2
            for (int k0 = 0; k0 < H; k0 += 16) {
                const int pb = (k0 >> 1) + hi;
                v2f av, bv;
                av = *(const v2f*)(arow + k0);      bv = bp[(pb + 0) * 32]; a0 = wmma4(av, bv, a0);
                av = *(const v2f*)(arow + k0 + 4);  bv = bp[(pb + 2) * 32]; a1 = wmma4(av, bv, a1);
                av = *(const v2f*)(arow + k0 + 8);  bv = bp[(pb + 4) * 32]; a2 = wmma4(av, bv, a2);
                av = *(const v2f*)(arow + k0 + 12); bv = bp[(pb + 6) * 32]; a3 = wmma4(av, bv, a3);
            }
            acc = (a0 + a1) + (a2 + a3);
        }

        // Fused epilogue: state = tanh(xw + h@W), written in place into out[t].
#pragma unroll
        for (int v = 0; v < 8; ++v) {
            const int m = v + 8 * hi;
            const size_t idx = (size_t)(row0 + m) * H + col0 + ln;
            xw[idx] = TANHF(xw[idx] + acc[v]);
        }

        // ---- Device-wide barrier (monotonic counter; no reset races).
        __syncthreads();
        if (threadIdx.x == 0) {
            __threadfence();  // release our state stores
            __hip_atomic_fetch_add(counter, 1u, __ATOMIC_RELEASE,
                                   __HIP_MEMORY_SCOPE_AGENT);
            const unsigned target = (unsigned)(t + 1) * REC_BLOCKS;
            while (__hip_atomic_load(counter, __ATOMIC_ACQUIRE,
                                     __HIP_MEMORY_SCOPE_AGENT) < target) {
                __builtin_amdgcn_s_sleep(8);
            }
        }
        __syncthreads();
        __builtin_amdgcn_fence(__ATOMIC_ACQUIRE, "agent");  // invalidate stale lines
    }
}

// ---------------------------------------------------------------------------
// Fallback step kernel (used only if ws_size can't hold the barrier counter).
// ---------------------------------------------------------------------------
__global__ __launch_bounds__(THREADS_PER_BLOCK)
void rnn_step_kernel(const float* __restrict__ S, const float* __restrict__ W,
                     float* __restrict__ XwOut, int N, int K) {
    const int lane = threadIdx.x & 31;
    const int wave = threadIdx.x >> 5;
    const int tileId = blockIdx.x * WAVES_PER_BLOCK + wave;
    const int tilesN = N >> 4;
    const int tm = tileId / tilesN;
    const int tn = tileId - tm * tilesN;

    const int row0 = tm << 4;
    const int col0 = tn << 4;
    const int ln = lane & 15;
    const int hi = lane >> 4;

    v8f acc = {0.f, 0.f, 0.f, 0.f, 0.f, 0.f, 0.f, 0.f};
    if (S != nullptr) {
        acc = wmma_tile_f32(S, W, row0, col0, K, N, lane);
    }

#pragma unroll
    for (int v = 0; v < 8; ++v) {
        const int m = v + 8 * hi;
        const size_t idx = (size_t)(row0 + m) * N + col0 + ln;
        XwOut[idx] = TANHF(XwOut[idx] + acc[v]);
    }
}

// ---------------------------------------------------------------------------
// Launch: inputs = {inputs[T,B,I], W_xh[I,H], W_hh[H,H], b_h[H]}
// d_out layout: outputs[T,B,H] then final_state[B,H] (flat, f32).
// ---------------------------------------------------------------------------
extern "C" void kernel_launch(void* const* d_in, const int* in_sizes, int n_in,
                              void* d_out, int out_size, void* d_ws, size_t ws_size,
                              hipStream_t stream) {
    (void)in_sizes; (void)n_in; (void)out_size;

    const float* X    = (const float*)d_in[0];  // [T, B, I]
    const float* W_xh = (const float*)d_in[1];  // [I, H]
    const float* W_hh = (const float*)d_in[2];  // [H, H]
    const float* b_h  = (const float*)d_in[3];  // [H]
    float* out = (float*)d_out;

    const int T = 512, B = 64, I = 1024, H = 1024;
    const int M = T * B;  // 32768

    // 1) Input projection for all timesteps: out[0:T*B*H) = X @ W_xh + b
    {
        const int tiles = (M / 32) * (H / 64);          // 16384 wave macro-tiles
        const int blocks = tiles / WAVES_PER_BLOCK;     // 2048
        xproj_kernel<<<blocks, THREADS_PER_BLOCK, 0, stream>>>(
            X, W_xh, b_h, out, M, H, I);
    }

    // 2) Recurrence: persistent kernel with device barrier (preferred), or
    //    512 sequential launches if no workspace for the barrier counter.
    if (ws_size >= 64) {
        (void)hipMemsetAsync(d_ws, 0, 64, stream);  // zero the barrier counter
        rnn_recurrence_kernel<<<REC_BLOCKS, THREADS_PER_BLOCK, 0, stream>>>(
            W_hh, out, (unsigned*)d_ws, T);
    } else {
        const int tiles = (B / 16) * (H / 16);          // 256
        const int blocks = tiles / WAVES_PER_BLOCK;     // 32
        for (int t = 0; t < T; ++t) {
            const float* S = (t == 0) ? nullptr : (out + (size_t)(t - 1) * B * H);
            rnn_step_kernel<<<blocks, THREADS_PER_BLOCK, 0, stream>>>(
                S, W_hh, out + (size_t)t * B * H, H, H);
        }
    }

    // 3) final_state = outputs[T-1]
    (void)hipMemcpyAsync(out + (size_t)T * B * H, out + (size_t)(T - 1) * B * H,
                         (size_t)B * H * sizeof(float), hipMemcpyDeviceToDevice, stream);
}